// backprop_35046933136092
// MI455X (gfx1250) — compile-verified
//
#include <hip/hip_runtime.h>
#include <math.h>

// ---------------- problem constants ----------------
#define IN_DIM    1024
#define HID       1024
#define OUT_DIM   64
#define N_SAMPLES 16384
#define LR        0.01f
#define EPS       1e-8f

// ---------------- launch geometry (wave32) ----------------
#define NBLK  64           // persistent workgroups (grid barrier members)
#define NTHR  256          // 8 waves of 32
#define WPB   8
#define NWAVE (NBLK * WPB) // 512 waves total

typedef __attribute__((ext_vector_type(2))) float v2f;
typedef __attribute__((ext_vector_type(8))) float v8f;

// ---------------- workspace layout (float offsets) ----------------
#define O_B1   0        // 1024  stored b1
#define O_B2   1024     // 1024  stored b2
#define O_B3   2048     // 64    stored b3
#define O_A1   2112     // 1024  activation a1
#define O_A2   3136     // 1024  activation a2
#define O_A3   4160     // 64    activation a3
#define O_D3   4224     // 64    delta d3
#define O_O2   4288     // 1024  delta o2
#define O_O1   5312     // 1024  delta o1
#define O_SCAL 6336     // 6 scales: s1,s2,s3,sb1,sb2,sb3
#define O_PART 6352     // 6*NBLK deterministic per-block sq partials
#define O_SYNC 6752     // 2 uints: generation, arrival count

__device__ __forceinline__ float wred32(float v) {
#pragma unroll
  for (int m = 16; m > 0; m >>= 1) v += __shfl_xor(v, m, 32);
  return v;
}

__device__ __forceinline__ float fast_sigmoid(float z) {
  return __builtin_amdgcn_rcpf(1.f + __expf(-z));   // v_rcp_f32, no div refinement
}

// grid-wide barrier: generation counter + arrival count in d_ws
__device__ __forceinline__ void gbar(unsigned* sync, unsigned& mygen) {
  __syncthreads();
  if (threadIdx.x == 0) {
    __threadfence();
    const unsigned g = mygen;
    if (atomicAdd(&sync[1], 1u) == NBLK - 1u) {
      sync[1] = 0u;
      __threadfence();
      atomicAdd(&sync[0], 1u);                 // release everyone
    } else {
      volatile unsigned* vg = &sync[0];
      while (*vg == g) __builtin_amdgcn_s_sleep(1);
    }
  }
  __syncthreads();
  __threadfence();                             // acquire for all waves in block
}

// barrier fused with scale recomputation: the LAST arriving block combines the
// per-block squared-norm partials (fixed summation order -> deterministic) and
// publishes the six normalization scales before releasing the grid.
__device__ __forceinline__ void gbar_scales(unsigned* sync, unsigned& mygen,
                                            const float* part, volatile float* scal) {
  __syncthreads();
  if (threadIdx.x == 0) {
    __threadfence();
    const unsigned g = mygen;
    if (atomicAdd(&sync[1], 1u) == NBLK - 1u) {
      __threadfence();
#pragma unroll
      for (int q = 0; q < 6; ++q) {
        float s = 0.f;
        for (int b = 0; b < NBLK; ++b) s += part[q * NBLK + b];
        scal[q] = 1.f / fmaxf(sqrtf(s), EPS);
      }
      sync[1] = 0u;
      __threadfence();
      atomicAdd(&sync[0], 1u);
    } else {
      volatile unsigned* vg = &sync[0];
      while (*vg == g) __builtin_amdgcn_s_sleep(1);
    }
  }
  __syncthreads();
  __threadfence();
}

// Forward matvec layer via V_WMMA_F32_16X16X4_F32.
// Input vector is staged once into LDS; block b owns output tile n0=b*16;
// its 8 waves split K=1024 (128 each, 32 WMMAs), partials reduced through LDS.
// A-matrix rows are replicated with the input vector so every row of D holds
// the same 16 partial sums.
__device__ __forceinline__ void fwd_layer(const float* W, float sw,
                                          const float* bst, float sb,
                                          const float* xv, float* act,
                                          float* svec, float* sred) {
  const int tid   = threadIdx.x;
  const int lane  = tid & 31;
  const int wv    = tid >> 5;
  const int n0    = blockIdx.x * 16;
  const int nlo   = lane & 15;
  const int khalf = (lane >= 16) ? 2 : 0;   // ISA: lanes 16-31 carry K=2,3
  ((float4*)svec)[tid] = ((const float4*)xv)[tid];     // stage x (1024 f32)
  __syncthreads();
  v8f c = {0.f, 0.f, 0.f, 0.f, 0.f, 0.f, 0.f, 0.f};
  const int kbase = wv * 128;
#pragma unroll 4
  for (int kk = 0; kk < 128; kk += 4) {
    const int k = kbase + kk + khalf;
    v2f a, b;
    a.x = svec[k];                           // broadcast from LDS
    a.y = svec[k + 1];
    const float* wp = W + (size_t)k * (size_t)HID + n0 + nlo;
    b.x = wp[0];                             // B[K,n] streamed from L2
    b.y = wp[HID];
    c = __builtin_amdgcn_wmma_f32_16x16x4_f32(false, a, false, b,
                                              (short)0, c, false, false);
  }
  if (lane < 16) sred[wv * 16 + lane] = c[0];   // row 0 of D
  __syncthreads();
  if (tid < 16) {
    float z = 0.f;
#pragma unroll
    for (int w = 0; w < WPB; ++w) z += sred[w * 16 + tid];
    z = sw * z + sb * bst[n0 + tid];
    act[n0 + tid] = fast_sigmoid(z);
  }
  __syncthreads();
}

__global__ void __launch_bounds__(NTHR)
train_kernel(const float* __restrict__ data, const float* __restrict__ tgts,
             float* wst, float* ws) {
  float* b1 = ws + O_B1;  float* b2 = ws + O_B2;  float* b3 = ws + O_B3;
  float* a1 = ws + O_A1;  float* a2 = ws + O_A2;  float* a3 = ws + O_A3;
  float* d3 = ws + O_D3;  float* o2 = ws + O_O2;  float* o1 = ws + O_O1;
  volatile float* scal = (volatile float*)(ws + O_SCAL);
  float* part = ws + O_PART;
  unsigned* sync = (unsigned*)(ws + O_SYNC);
  float* w1 = wst;
  float* w2 = wst + (size_t)HID * HID;
  float* w3 = wst + 2ull * HID * HID;

  __shared__ float svec[1024];   // staged vector (x / a1 / a2 / d3 / o2 / o1)
  __shared__ float sred[256];    // reduction scratch
  unsigned mygen = 0;
  const int tid   = threadIdx.x;
  const int lane  = tid & 31;
  const int wv    = tid >> 5;
  const int nlo   = lane & 15;
  const int khalf = (lane >= 16) ? 2 : 0;
  const int gw    = blockIdx.x * WPB + wv;   // 0..511

  for (int t = 0; t < N_SAMPLES; ++t) {
    const float* x  = data + (size_t)t * IN_DIM;
    const float* tg = tgts + (size_t)t * OUT_DIM;
    const float s1 = scal[0], s2 = scal[1], s3 = scal[2];
    const float sb1 = scal[3], sb2 = scal[4], sb3 = scal[5];
    if (tid == 0 && t + 1 < N_SAMPLES)
      __builtin_prefetch(data + (size_t)(t + 1) * IN_DIM, 0, 1);

    // -------- Phase A: a1 = sigmoid(s1*(x@w1) + sb1*b1) --------
    fwd_layer(w1, s1, b1, sb1, x, a1, svec, sred);
    gbar(sync, mygen);

    // -------- Phase B: a2 = sigmoid(s2*(a1@w2) + sb2*b2) --------
    fwd_layer(w2, s2, b2, sb2, a1, a2, svec, sred);
    gbar(sync, mygen);

    // -------- Phase C (block 0): layer3 via WMMA + softmax + d3 + b3 ----
    if (blockIdx.x == 0) {
      ((float4*)svec)[tid] = ((const float4*)a2)[tid];   // stage a2
      __syncthreads();
      // 4 output tiles of 16; 2 waves per tile split K=1024 in halves
      const int tile  = wv >> 1;
      const int n0t   = tile * 16;
      const int kbase = (wv & 1) * 512;
      v8f c = {0.f, 0.f, 0.f, 0.f, 0.f, 0.f, 0.f, 0.f};
#pragma unroll 4
      for (int kk = 0; kk < 512; kk += 4) {
        const int k = kbase + kk + khalf;
        v2f a, b;
        a.x = svec[k];
        a.y = svec[k + 1];
        const float* wp = w3 + (size_t)k * OUT_DIM + n0t + nlo;
        b.x = wp[0];
        b.y = wp[OUT_DIM];
        c = __builtin_amdgcn_wmma_f32_16x16x4_f32(false, a, false, b,
                                                  (short)0, c, false, false);
      }
      if (lane < 16) sred[wv * 16 + lane] = c[0];
      __syncthreads();
      float a3v = 0.f;
      if (tid < OUT_DIM) {
        const int tl = tid >> 4, nn = tid & 15;
        float z = s3 * (sred[(2 * tl) * 16 + nn] + sred[(2 * tl + 1) * 16 + nn])
                + sb3 * b3[tid];
        a3v = fast_sigmoid(z);
        a3[tid] = a3v;
        sred[128 + tid] = __expf(-a3v);                  // e
      }
      __syncthreads();
      if (tid == 0) {
        float s = 0.f;
        for (int i = 0; i < 64; ++i) s += sred[128 + i];
        sred[192] = __builtin_amdgcn_rcpf(s);            // 1/sum(e)
      }
      __syncthreads();
      if (tid < OUT_DIM) {
        const float net = sred[128 + tid] * sred[192];
        const float d = (tg[tid] - net) * a3v * (1.f - a3v);
        d3[tid] = d;
        const float bn = sb3 * b3[tid] - LR * d;         // effective update
        b3[tid] = bn;
        sred[tid] = bn * bn;
      }
      __syncthreads();
      if (tid == 0) {
        float s = 0.f; for (int i = 0; i < 64; ++i) s += sred[i];
        part[5 * NBLK + 0] = s;
      }
    } else if (tid == 0) {
      part[5 * NBLK + blockIdx.x] = 0.f;
    }
    gbar(sync, mygen);

    // -------- Phase D: o2 = s3*(w3_old@d3)*da2 ; w3 <- s3*w3 - LR*outer(a2,d3)
    {
      if (tid < 16) ((float4*)svec)[tid] = ((const float4*)d3)[tid]; // stage d3
      __syncthreads();
      float sqp = 0.f;
      const float2 dv = ((const float2*)svec)[lane];
#pragma unroll
      for (int rr = 0; rr < 2; ++rr) {
        const int r = gw + rr * NWAVE;               // row of w3 (1024 rows)
        const float a2r = a2[r];
        float2* wrow = (float2*)(w3 + (size_t)r * OUT_DIM);
        float2 wv2 = wrow[lane];
        const float dotp = wv2.x * dv.x + wv2.y * dv.y;
        float2 nw;
        nw.x = s3 * wv2.x - LR * a2r * dv.x;
        nw.y = s3 * wv2.y - LR * a2r * dv.y;
        wrow[lane] = nw;
        sqp += nw.x * nw.x + nw.y * nw.y;
        const float ds = wred32(dotp);
        if (lane == 0) o2[r] = s3 * ds * a2r * (1.f - a2r);
      }
      const float wsum = wred32(sqp);
      if (lane == 0) sred[wv] = wsum;
      __syncthreads();
      if (tid == 0) {
        float s = 0.f; for (int w = 0; w < WPB; ++w) s += sred[w];
        part[2 * NBLK + blockIdx.x] = s;
      }
    }
    gbar(sync, mygen);

    // -------- Phase E: o1 = s2*(w2_old@o2)*da1 ; w2 <- s2*w2 - LR*outer(a1,o2)
    //                   b2 <- sb2*b2 - LR*o2
    {
      ((float4*)svec)[tid] = ((const float4*)o2)[tid];   // stage o2
      __syncthreads();
      float sqp = 0.f;
      const float4* ovv = (const float4*)svec;
#pragma unroll
      for (int rr = 0; rr < 2; ++rr) {
        const int r = gw + rr * NWAVE;
        const float a1r = a1[r];
        float4* wrow = (float4*)(w2 + (size_t)r * HID);
        float dotp = 0.f;
#pragma unroll
        for (int c4 = 0; c4 < 8; ++c4) {
          const int idx = c4 * 32 + lane;
          float4 wv4 = wrow[idx];
          const float4 ov = ovv[idx];
          dotp += wv4.x * ov.x + wv4.y * ov.y + wv4.z * ov.z + wv4.w * ov.w;
          float4 nw;
          nw.x = s2 * wv4.x - LR * a1r * ov.x;
          nw.y = s2 * wv4.y - LR * a1r * ov.y;
          nw.z = s2 * wv4.z - LR * a1r * ov.z;
          nw.w = s2 * wv4.w - LR * a1r * ov.w;
          wrow[idx] = nw;
          sqp += nw.x * nw.x + nw.y * nw.y + nw.z * nw.z + nw.w * nw.w;
        }
        const float ds = wred32(dotp);
        if (lane == 0) o1[r] = s2 * ds * a1r * (1.f - a1r);
      }
      const float wsum = wred32(sqp);
      if (lane == 0) sred[wv] = wsum;
      if (tid < 16) {
        const int j = blockIdx.x * 16 + tid;
        const float bn = sb2 * b2[j] - LR * svec[j];
        b2[j] = bn;
        sred[16 + tid] = bn * bn;
      }
      __syncthreads();
      if (tid == 0) {
        float s = 0.f; for (int w = 0; w < WPB; ++w) s += sred[w];
        part[1 * NBLK + blockIdx.x] = s;
        float sbp = 0.f; for (int i = 0; i < 16; ++i) sbp += sred[16 + i];
        part[4 * NBLK + blockIdx.x] = sbp;
      }
    }
    gbar(sync, mygen);

    // -------- Phase F: w1 <- s1*w1 - LR*outer(x,o1) ; b1 <- sb1*b1 - LR*o1
    //          barrier fused with scale recomputation (phase G)
    {
      ((float4*)svec)[tid] = ((const float4*)o1)[tid];   // stage o1
      __syncthreads();
      float sqp = 0.f;
      const float4* ovv = (const float4*)svec;
#pragma unroll
      for (int rr = 0; rr < 2; ++rr) {
        const int r = gw + rr * NWAVE;
        const float xr = x[r];
        float4* wrow = (float4*)(w1 + (size_t)r * HID);
#pragma unroll
        for (int c4 = 0; c4 < 8; ++c4) {
          const int idx = c4 * 32 + lane;
          float4 wv4 = wrow[idx];
          const float4 ov = ovv[idx];
          float4 nw;
          nw.x = s1 * wv4.x - LR * xr * ov.x;
          nw.y = s1 * wv4.y - LR * xr * ov.y;
          nw.z = s1 * wv4.z - LR * xr * ov.z;
          nw.w = s1 * wv4.w - LR * xr * ov.w;
          wrow[idx] = nw;
          sqp += nw.x * nw.x + nw.y * nw.y + nw.z * nw.z + nw.w * nw.w;
        }
      }
      const float wsum = wred32(sqp);
      if (lane == 0) sred[wv] = wsum;
      if (tid < 16) {
        const int j = blockIdx.x * 16 + tid;
        const float bn = sb1 * b1[j] - LR * svec[j];
        b1[j] = bn;
        sred[16 + tid] = bn * bn;
      }
      __syncthreads();
      if (tid == 0) {
        float s = 0.f; for (int w = 0; w < WPB; ++w) s += sred[w];
        part[0 * NBLK + blockIdx.x] = s;
        float sbp = 0.f; for (int i = 0; i < 16; ++i) sbp += sred[16 + i];
        part[3 * NBLK + blockIdx.x] = sbp;
      }
    }
    gbar_scales(sync, mygen, part, scal);   // fused phase G
  }

  // -------- materialize d_out = scale * stored --------
  {
    const float s1 = scal[0], s2 = scal[1], s3 = scal[2];
    const size_t gsz = (size_t)NBLK * NTHR;
    const size_t g = (size_t)blockIdx.x * NTHR + tid;
    for (size_t i = g; i < (size_t)HID * HID; i += gsz) w1[i] *= s1;
    for (size_t i = g; i < (size_t)HID * HID; i += gsz) w2[i] *= s2;
    for (size_t i = g; i < (size_t)HID * OUT_DIM; i += gsz) w3[i] *= s3;
  }
}

__global__ void init_kernel(const float* b1i, const float* b2i, const float* b3i,
                            float* ws) {
  const int g = blockIdx.x * blockDim.x + threadIdx.x;
  const int n = gridDim.x * blockDim.x;
  for (int i = g; i < 1024; i += n) {
    ws[O_B1 + i] = b1i[i];
    ws[O_B2 + i] = b2i[i];
    ws[O_A1 + i] = 0.f; ws[O_A2 + i] = 0.f;
    ws[O_O1 + i] = 0.f; ws[O_O2 + i] = 0.f;
  }
  for (int i = g; i < 64; i += n) {
    ws[O_B3 + i] = b3i[i];
    ws[O_A3 + i] = 0.f; ws[O_D3 + i] = 0.f;
  }
  for (int i = g; i < 6; i += n) ws[O_SCAL + i] = 1.f;
  for (int i = g; i < 6 * NBLK; i += n) ws[O_PART + i] = 0.f;
  if (g < 2) ((unsigned*)(ws + O_SYNC))[g] = 0u;
}

__global__ void copyw_kernel(const float* w1i, const float* w2i, const float* w3i,
                             float* out) {
  const size_t g = (size_t)blockIdx.x * blockDim.x + threadIdx.x;
  const size_t n = (size_t)gridDim.x * blockDim.x;
  for (size_t i = g; i < (size_t)HID * HID; i += n) {
    out[i] = w1i[i];
    out[(size_t)HID * HID + i] = w2i[i];
  }
  for (size_t i = g; i < (size_t)HID * OUT_DIM; i += n)
    out[2ull * HID * HID + i] = w3i[i];
}

extern "C" void kernel_launch(void* const* d_in, const int* in_sizes, int n_in,
                              void* d_out, int out_size, void* d_ws, size_t ws_size,
                              hipStream_t stream) {
  (void)in_sizes; (void)n_in; (void)out_size; (void)ws_size;
  const float* data = (const float*)d_in[0];
  const float* tgts = (const float*)d_in[1];
  const float* w1i  = (const float*)d_in[2];
  const float* b1i  = (const float*)d_in[3];
  const float* w2i  = (const float*)d_in[4];
  const float* b2i  = (const float*)d_in[5];
  const float* w3i  = (const float*)d_in[6];
  const float* b3i  = (const float*)d_in[7];
  float* out = (float*)d_out;
  float* ws  = (float*)d_ws;

  init_kernel<<<8, 256, 0, stream>>>(b1i, b2i, b3i, ws);
  copyw_kernel<<<512, 256, 0, stream>>>(w1i, w2i, w3i, out);
  train_kernel<<<NBLK, NTHR, 0, stream>>>(data, tgts, out, ws);
}